// AttentionLoss_63299228008648
// MI455X (gfx1250) — compile-verified
//
#include <hip/hip_runtime.h>

// Problem constants (fixed by the reference setup)
#define BS  2
#define H   64
#define W   64
#define HW  (H * W)        // 4096
#define ROW (64 * 64)      // mh*mw elements per (b,p) row of sim

typedef __attribute__((ext_vector_type(2))) float v2f;
typedef __attribute__((ext_vector_type(8))) float v8f;

// Reduce 32 lane-partials to a full wave sum using the matrix pipe:
// A = ones(16x4), B lane L: v0 = x_L, v1 = 0  =>  D[m][n] = x_n + x_{n+16}
// (replicated across rows). Then 4 xor-shuffles sum the 16 column sums.
// Requires EXEC == all ones (call only from fully-active waves).
__device__ __forceinline__ float wave_sum32(float x) {
    v2f a; a[0] = 1.0f; a[1] = 1.0f;
    v2f b; b[0] = x;    b[1] = 0.0f;
    v8f c = {};
    c = __builtin_amdgcn_wmma_f32_16x16x4_f32(false, a, false, b,
                                              (short)0, c, false, false);
    float s = c[0];
    s += __shfl_xor(s, 1, 32);
    s += __shfl_xor(s, 2, 32);
    s += __shfl_xor(s, 4, 32);
    s += __shfl_xor(s, 8, 32);
    return s;   // full 32-lane sum in every lane
}

__global__ void init_ws_kernel(double* ws) {
    if (threadIdx.x < 3) ws[threadIdx.x] = 0.0;
}

// One block per (b, p). 256 threads, each does 4 coalesced float4 loads.
__global__ void cwg_kernel(const float* __restrict__ sim,
                           const float* __restrict__ wc,
                           const int*   __restrict__ mask,
                           double*      __restrict__ ws) {
    const int bp = blockIdx.x;                 // b*HW + p
    if (mask[bp] == 0) return;                 // masked row: skip 16KB of traffic
    const float wcy = wc[2 * bp + 0];
    const float wcx = wc[2 * bp + 1];
    const float4* row = reinterpret_cast<const float4*>(sim) + (size_t)bp * (ROW / 4);
    const int t = threadIdx.x;                 // 0..255
    float acc = 0.0f;
#pragma unroll
    for (int i = 0; i < 4; ++i) {
        const int e0 = i * 1024 + t * 4;       // element index, 16B aligned
        const float4 s4 = row[i * 256 + t];    // fully coalesced per wave
        const float dy  = (float)(e0 >> 6) - wcy;
        const float dy2 = dy * dy;
        const float fx0 = (float)(e0 & 63);
        const float vals[4] = {s4.x, s4.y, s4.z, s4.w};
#pragma unroll
        for (int j = 0; j < 4; ++j) {
            const float dx   = (fx0 + (float)j) - wcx;
            const float dist = __builtin_amdgcn_sqrtf(dy2 + dx * dx);
            // exp(-dist/2) = exp2(dist * -0.5*log2(e))
            acc += vals[j] * __builtin_amdgcn_exp2f(dist * -0.72134752044448f);
        }
    }
    const float wsum = wave_sum32(acc);
    __shared__ float smem[8];
    const int lane = t & 31, wv = t >> 5;
    if (lane == 0) smem[wv] = wsum;
    __syncthreads();
    if (t == 0) {
        float tot = 0.0f;
#pragma unroll
        for (int w = 0; w < 8; ++w) tot += smem[w];
        atomicAdd(&ws[0], (double)tot);        // global_atomic_add_f64
    }
}

// TV: 16128 masked neighbor pairs (8064 y-dir + 8064 x-dir), both channels each.
// Grid sized so every lane is active (WMMA EXEC requirement).
__global__ void tv_kernel(const float* __restrict__ wc,
                          const int*   __restrict__ mask,
                          double*      __restrict__ ws) {
    const int idx = blockIdx.x * 256 + threadIdx.x;   // 0..16127 exactly
    float contrib = 0.0f;
    int g0, g1;
    if (idx < 8064) {                                  // y-direction pairs
        const int b = idx / 4032, r = idx % 4032;
        const int h = r >> 6, w = r & 63;              // h in 0..62
        g0 = b * HW + h * W + w; g1 = g0 + W;
    } else {                                           // x-direction pairs
        const int r2 = idx - 8064;
        const int b = r2 / 4032, r = r2 % 4032;
        const int h = r / 63, w = r % 63;              // w in 0..62
        g0 = b * HW + h * W + w; g1 = g0 + 1;
    }
    if (mask[g0] && mask[g1]) {
        const float d0 = wc[2 * g1 + 0] - wc[2 * g0 + 0];
        const float d1 = wc[2 * g1 + 1] - wc[2 * g0 + 1];
        contrib = d0 * d0 + d1 * d1;
    }
    const float s = wave_sum32(contrib);
    if ((threadIdx.x & 31) == 0) atomicAdd(&ws[1], (double)s);
}

// DCML: masks make it per-line pair sums. Block per (b, row|col, line), 64 threads.
// rows: relu(x[q]-x[p]) for same row, j_q>j_p ; cols: relu(y[q]-y[p]), i_q>i_p.
__global__ void dcml_kernel(const float* __restrict__ wc,
                            const int*   __restrict__ mask,
                            double*      __restrict__ ws) {
    __shared__ float sv[64];
    __shared__ float sm[64];
    __shared__ float part[2];
    const int d    = blockIdx.x;          // 0..255
    const int b    = d >> 7;
    const int mode = (d >> 6) & 1;        // 0 = row (x-channel), 1 = col (y-channel)
    const int line = d & 63;
    const int t    = threadIdx.x;         // 0..63
    const int p    = (mode == 0) ? (line * W + t) : (t * W + line);
    const int g    = b * HW + p;
    sv[t] = (mode == 0) ? wc[2 * g + 1] : wc[2 * g + 0];
    sm[t] = mask[g] ? 1.0f : 0.0f;
    __syncthreads();
    float s = 0.0f;
    const float vt = sv[t];
    for (int u = t + 1; u < 64; ++u)
        s += fmaxf(sv[u] - vt, 0.0f) * sm[u];
    s *= sm[t];
    const float wsum = wave_sum32(s);     // two fully-active waves
    if ((t & 31) == 0) part[t >> 5] = wsum;
    __syncthreads();
    if (t == 0) atomicAdd(&ws[2], (double)(part[0] + part[1]));
}

__global__ void finalize_kernel(const double* __restrict__ ws,
                                float* __restrict__ out) {
    const double cwg  = -2.0   * ws[0] / 33554432.0;  // mean over BS*HW*mh*mw
    const double tv   =  1e-4  * ws[1] / 16128.0;     // y.mean() + x.mean(), equal counts
    const double dcml = -0.01  * ws[2] / 33554432.0;  // mean over BS*HW*HW
    out[0] = (float)(cwg + tv + dcml);
}

extern "C" void kernel_launch(void* const* d_in, const int* in_sizes, int n_in,
                              void* d_out, int out_size, void* d_ws, size_t ws_size,
                              hipStream_t stream) {
    (void)in_sizes; (void)n_in; (void)out_size; (void)ws_size;
    const float* sim  = (const float*)d_in[0];   // [2,4096,64,64] f32
    const float* wc   = (const float*)d_in[1];   // [2,4096,2]     f32
    const int*   mask = (const int*)d_in[2];     // [2,64,64]      int (bool)
    double* ws  = (double*)d_ws;
    float*  out = (float*)d_out;

    hipLaunchKernelGGL(init_ws_kernel, dim3(1),       dim3(32),  0, stream, ws);
    hipLaunchKernelGGL(cwg_kernel,     dim3(BS * HW), dim3(256), 0, stream, sim, wc, mask, ws);
    hipLaunchKernelGGL(tv_kernel,      dim3(63),      dim3(256), 0, stream, wc, mask, ws);
    hipLaunchKernelGGL(dcml_kernel,    dim3(256),     dim3(64),  0, stream, wc, mask, ws);
    hipLaunchKernelGGL(finalize_kernel, dim3(1),      dim3(1),   0, stream, ws, out);
}